// BayerDenoise_7181185319108
// MI455X (gfx1250) — compile-verified
//
#include <hip/hip_runtime.h>

// ---------------- problem constants ----------------
#define H   4096
#define W   6144
#define HC  2048      // per-channel height
#define WC  3072      // per-channel width
#define TS  32        // per-channel output tile
#define EXT 36        // TS + 2*R   (a/b extended extent)
#define PL  40        // TS + 4*R   (I plane extent)
#define PLS 41        // padded plane stride
#define RAWN 80       // 2*PL  (full-res staging tile)
#define OT  64        // 2*TS  (full-res output tile)

// ---------------- CDNA5 async global<->LDS paths ----------------
#if defined(__has_builtin)
#if __has_builtin(__builtin_amdgcn_global_load_async_to_lds_b128)
#define USE_ASYNC_LD 1
#endif
#if __has_builtin(__builtin_amdgcn_global_store_async_from_lds_b128)
#define USE_ASYNC_ST 1
#endif
#endif
#ifndef USE_ASYNC_LD
#define USE_ASYNC_LD 0
#endif
#ifndef USE_ASYNC_ST
#define USE_ASYNC_ST 0
#endif

// 16-byte int vector, matching the builtin's parameter type
typedef int v4i __attribute__((__vector_size__(4 * sizeof(int))));
typedef __attribute__((address_space(1))) v4i gv4i;   // global (AS1)
typedef __attribute__((address_space(3))) v4i lv4i;   // LDS    (AS3)

__device__ __forceinline__ void async_copy16_in(const float* gsrc, float* ldst) {
#if USE_ASYNC_LD
  __builtin_amdgcn_global_load_async_to_lds_b128(
      (gv4i*)gsrc, (lv4i*)ldst, 0, 0);
#else
  *(float4*)ldst = *(const float4*)gsrc;
#endif
}

__device__ __forceinline__ void async_copy16_out(float* gdst, const float* lsrc) {
#if USE_ASYNC_ST
  __builtin_amdgcn_global_store_async_from_lds_b128(
      (gv4i*)gdst, (lv4i*)lsrc, 0, 0);
#else
  *(float4*)gdst = *(const float4*)lsrc;
#endif
}

__device__ __forceinline__ void wait_async_done() {
#if USE_ASYNC_LD
#if __has_builtin(__builtin_amdgcn_s_wait_asynccnt)
  __builtin_amdgcn_s_wait_asynccnt(0);
#else
  asm volatile("s_wait_asynccnt 0x0" ::: "memory");
#endif
#endif
}

// jnp.pad 'reflect' (edge not repeated); radius<=4, n>=8 -> single fold is exact
__device__ __forceinline__ int refl(int i, int n) {
  i = (i < 0) ? -i : i;
  i = (i >= n) ? (2 * n - 2 - i) : i;
  return i;
}

__global__ __launch_bounds__(256)
void bayer_guided_denoise(const float* __restrict__ x,
                          const float* __restrict__ bk,
                          const float* __restrict__ epsp,
                          float* __restrict__ out) {
  // raw staging buffer; after de-interleave it is reused as S/T scratch
  __shared__ float raw[RAWN * RAWN];            // 6400 f  (25.6 KB)
  __shared__ float plane[4][PL * PLS];          // 6560 f  (26.2 KB)
  __shared__ float aA[EXT * EXT];               // 1296 f
  __shared__ float bB[EXT * EXT];               // 1296 f
  __shared__ float otile[OT * OT];              // 4096 f  (16.4 KB) -> 78.6 KB total

  float* S_I  = raw;                  // PL*EXT = 1440
  float* S_II = raw + 1440;           // 1440
  float* T_a  = raw + 2880;           // EXT*TS = 1152
  float* T_b  = raw + 4032;           // 1152   (<= 6400 total, fits in raw)

  const int tid = threadIdx.x;
  const int tx0 = blockIdx.x * TS;    // channel-domain tile origin
  const int ty0 = blockIdx.y * TS;
  const float w   = bk[0];            // uniform 1/25 box weight
  const float eps = epsp[0];

  // ---- Phase 1: async-stage 80x80 full-res region into LDS ----
  const int fy0 = 2 * ty0 - 8;
  const int fx0 = 2 * tx0 - 8;        // 16B aligned: 4*fx0 = 256*bx - 32
  for (int c = tid; c < (RAWN * RAWN / 4); c += 256) {
    const int ry = c / (RAWN / 4);
    const int rx = (c % (RAWN / 4)) * 4;
    const int fy = fy0 + ry;
    const int fx = fx0 + rx;
    // whole 16B chunk is either fully inside the image or fully outside;
    // outside chunks are never read back (reflect folds into valid range)
    if (fy >= 0 && fy < H && fx >= 0 && fx + 3 < W) {
      async_copy16_in(x + (size_t)fy * W + fx, &raw[ry * RAWN + rx]);
    }
  }
  wait_async_done();
  __syncthreads();

  // ---- Phase 2: de-interleave Bayer quad into 4 channel planes ----
  for (int i = tid; i < PL * PL; i += 256) {
    const int py = i / PL;
    const int px = i % PL;
    const float v00 = raw[(2 * py)     * RAWN + 2 * px];
    const float v01 = raw[(2 * py)     * RAWN + 2 * px + 1];
    const float v10 = raw[(2 * py + 1) * RAWN + 2 * px];
    const float v11 = raw[(2 * py + 1) * RAWN + 2 * px + 1];
    plane[0][py * PLS + px] = v00;   // r
    plane[1][py * PLS + px] = v01;   // gr
    plane[2][py * PLS + px] = v10;   // gb
    plane[3][py * PLS + px] = v11;   // b
  }
  __syncthreads();

  // ---- Phase 3: per-channel separable guided filter ----
  for (int ch = 0; ch < 4; ++ch) {
    const float* P = plane[ch];

    // Pass A: horizontal 5-tap sums of I and I*I (reflect in x)
    for (int i = tid; i < PL * EXT; i += 256) {
      const int py = i / EXT;             // plane row (gy = ty0-4+py)
      const int sx = i % EXT;             // gx = tx0-2+sx
      const int gy = ty0 - 4 + py;
      const int gx = tx0 - 2 + sx;
      if (gy >= 0 && gy < HC && gx >= 0 && gx < WC) {
        float sI = 0.f, sII = 0.f;
#pragma unroll
        for (int dx = -2; dx <= 2; ++dx) {
          const int px = refl(gx + dx, WC) - (tx0 - 4);
          const float v = P[py * PLS + px];
          sI += v;
          sII += v * v;
        }
        S_I [py * EXT + sx] = sI;
        S_II[py * EXT + sx] = sII;
      }
    }
    __syncthreads();

    // Pass B: vertical 5-tap (reflect in y) + pointwise a,b
    for (int i = tid; i < EXT * EXT; i += 256) {
      const int ay = i / EXT;             // gy = ty0-2+ay
      const int sx = i % EXT;
      const int gy = ty0 - 2 + ay;
      const int gx = tx0 - 2 + sx;
      if (gy >= 0 && gy < HC && gx >= 0 && gx < WC) {
        float sI = 0.f, sII = 0.f;
#pragma unroll
        for (int dy = -2; dy <= 2; ++dy) {
          const int py = refl(gy + dy, HC) - (ty0 - 4);
          sI  += S_I [py * EXT + sx];
          sII += S_II[py * EXT + sx];
        }
        const float mI  = sI * w;
        const float mII = sII * w;
        const float var = mII - mI * mI;
        const float a   = var / (var + eps);
        aA[ay * EXT + sx] = a;
        bB[ay * EXT + sx] = mI - a * mI;
      }
    }
    __syncthreads();

    // Pass C: horizontal 5-tap sums of a,b (reflect in x on channel domain)
    for (int i = tid; i < EXT * TS; i += 256) {
      const int ay = i / TS;
      const int ox = i % TS;
      const int gy = ty0 - 2 + ay;
      if (gy >= 0 && gy < HC) {
        const int gx = tx0 + ox;
        float sa = 0.f, sb = 0.f;
#pragma unroll
        for (int dx = -2; dx <= 2; ++dx) {
          const int sx = refl(gx + dx, WC) - (tx0 - 2);   // always in [0,EXT)
          sa += aA[ay * EXT + sx];
          sb += bB[ay * EXT + sx];
        }
        T_a[ay * TS + ox] = sa;
        T_b[ay * TS + ox] = sb;
      }
    }
    __syncthreads();

    // Pass D: vertical 5-tap (reflect in y) -> interleaved output tile in LDS
    for (int i = tid; i < TS * TS; i += 256) {
      const int oy = i / TS;
      const int ox = i % TS;
      const int gy = ty0 + oy;
      const int gx = tx0 + ox;
      float sa = 0.f, sb = 0.f;
#pragma unroll
      for (int dy = -2; dy <= 2; ++dy) {
        const int ay = refl(gy + dy, HC) - (ty0 - 2);      // always in [0,EXT)
        sa += T_a[ay * TS + ox];
        sb += T_b[ay * TS + ox];
      }
      const float ma = sa * w;
      const float mb = sb * w;
      const float Iv = P[(oy + 4) * PLS + (ox + 4)];
      otile[(2 * oy + (ch >> 1)) * OT + 2 * ox + (ch & 1)] = ma * Iv + mb;
    }
    // no barrier needed: next Pass A writes S (disjoint from T/plane read by D)
  }
  __syncthreads();   // all channels assembled into otile

  // ---- Phase 4: async-flush interleaved 64x64 full-res tile, coalesced 16B ----
  const int oy0 = 2 * ty0;            // full-res tile origin (always in-range)
  const int ox0 = 2 * tx0;
  for (int c = tid; c < (OT * OT / 4); c += 256) {
    const int row = c / (OT / 4);
    const int col = (c % (OT / 4)) * 4;
    async_copy16_out(out + (size_t)(oy0 + row) * W + (ox0 + col),
                     &otile[row * OT + col]);
  }
  // S_ENDPGM performs an implicit wait-idle, covering outstanding ASYNCcnt.
}

extern "C" void kernel_launch(void* const* d_in, const int* in_sizes, int n_in,
                              void* d_out, int out_size, void* d_ws, size_t ws_size,
                              hipStream_t stream) {
  (void)in_sizes; (void)n_in; (void)out_size; (void)d_ws; (void)ws_size;
  const float* x    = (const float*)d_in[0];
  const float* bk   = (const float*)d_in[1];
  const float* eps  = (const float*)d_in[2];
  float* out        = (float*)d_out;

  dim3 grid(WC / TS, HC / TS);   // 96 x 64 tiles
  bayer_guided_denoise<<<grid, dim3(256), 0, stream>>>(x, bk, eps, out);
}